// CoulombDSF_NB_47991964566175
// MI455X (gfx1250) — compile-verified
//
#include <hip/hip_runtime.h>
#include <hip/hip_bf16.h>
#include <stdint.h>

// CoulombDSF neighbor-list energy for MI455X (gfx1250, wave32).
// Wave-per-atom, 2 neighbors/lane; neighbor-index rows staged to LDS via the
// CDNA5 async global->LDS DMA path (ASYNCcnt), double-buffered.

#define N_NB            64
#define WAVES_PER_BLOCK 8
#define BLOCK_THREADS   256
#define ATOMS_PER_WAVE  8

#define ALPHA_F        0.2f
#define COUL_CONST_F   7.1998226f
#define LOG2E_F        1.44269504088896340736f
#define TWO_A_ISQRTPI  0.22567583341910251f   // 2*ALPHA/sqrt(pi)

// ---------------- async global->LDS (CDNA5 ASYNCcnt path) ----------------
#if __has_builtin(__builtin_amdgcn_global_load_async_to_lds_b64)
  #define ASYNC_TIER 1
#else
  #define ASYNC_TIER 2   // inline-asm fallback per CDNA5 ISA
#endif

typedef int v2i __attribute__((vector_size(8)));              // 2 x i32
typedef __attribute__((address_space(1))) v2i gas_v2i;        // global
typedef __attribute__((address_space(3))) v2i las_v2i;        // LDS
typedef __attribute__((address_space(3))) void las_void;

__device__ __forceinline__ void async_idx_row_to_lds(const int* gsrc, void* lds_dst) {
#if ASYNC_TIER == 1
  __builtin_amdgcn_global_load_async_to_lds_b64((gas_v2i*)gsrc, (las_v2i*)lds_dst,
                                                /*offset=*/0, /*cpol=*/0);
#else
  unsigned loff = (unsigned)(uintptr_t)(las_void*)lds_dst;  // LDS byte offset
  asm volatile("global_load_async_to_lds_b64 %0, %1, off"
               :
               : "v"(loff), "v"(gsrc)
               : "memory");
#endif
}

#if __has_builtin(__builtin_amdgcn_s_wait_asynccnt)
  #define WAIT_ASYNCCNT(n) __builtin_amdgcn_s_wait_asynccnt(n)
#else
  #define WAIT_ASYNCCNT(n) asm volatile("s_wait_asynccnt %0" :: "n"(n) : "memory")
#endif

// ---------------- fast erfc: Abramowitz & Stegun 7.1.26, |err|<=1.5e-7 -----
__device__ __forceinline__ float erfc_fast(float x) {
  float t = __builtin_amdgcn_rcpf(fmaf(0.3275911f, x, 1.0f));
  float p = fmaf(1.061405429f, t, -1.453152027f);
  p = fmaf(p, t, 1.421413741f);
  p = fmaf(p, t, -0.284496736f);
  p = fmaf(p, t, 0.254829592f);
  p *= t;
  return p * __builtin_amdgcn_exp2f(-(x * x) * LOG2E_F);  // exp(-x^2)
}

__global__ __launch_bounds__(BLOCK_THREADS) void coul_dsf_kernel(
    const float* __restrict__ coord,
    const float* __restrict__ charges,
    const int*   __restrict__ idxj,
    const unsigned char* __restrict__ padmask,
    const float* __restrict__ cutoff_p,
    float* __restrict__ out,
    int n_atoms, int waves_total)
{
  __shared__ __align__(16) int lds_idx[WAVES_PER_BLOCK][2][N_NB];

  const int lane = threadIdx.x & 31;
  const int wid  = threadIdx.x >> 5;
  int a = blockIdx.x * WAVES_PER_BLOCK + wid;

  // shifted-force constants (wave-uniform, computed once)
  const float Rc    = cutoff_p[0];
  const float invRc = __builtin_amdgcn_rcpf(Rc);
  const float aRc   = ALPHA_F * Rc;
  const float c2    = erfc_fast(aRc) * invRc;
  const float c4    = TWO_A_ISQRTPI * __builtin_amdgcn_exp2f(-(aRc * aRc) * LOG2E_F) * invRc;
  const float c34   = fmaf(c2, invRc, c4);  // c3 + c4

  // prime the double buffer: each lane fetches its 8B of the idx row async
  if (a < n_atoms) {
    async_idx_row_to_lds(idxj + (size_t)a * N_NB + 2 * lane,
                         &lds_idx[wid][0][2 * lane]);
  }

  int k = 0;
  for (; a < n_atoms; a += waves_total, ++k) {
    const int cur   = k & 1;
    const int anext = a + waves_total;
    if (anext < n_atoms) {
      async_idx_row_to_lds(idxj + (size_t)anext * N_NB + 2 * lane,
                           &lds_idx[wid][cur ^ 1][2 * lane]);
      WAIT_ASYNCCNT(1);  // async loads complete in order -> current buffer ready
    } else {
      WAIT_ASYNCCNT(0);
    }
    asm volatile("" ::: "memory");  // don't hoist the LDS read above the wait

    const int2 jj = *(const int2*)&lds_idx[wid][cur][2 * lane];

    // atom-i data: wave-uniform -> force scalar path
    const int   ua = __builtin_amdgcn_readfirstlane(a);
    const float xi = coord[(size_t)3 * ua + 0];
    const float yi = coord[(size_t)3 * ua + 1];
    const float zi = coord[(size_t)3 * ua + 2];
    const float qi = charges[ua];

    const unsigned short mm =
        *(const unsigned short*)(padmask + (size_t)a * N_NB + 2 * lane);

    float acc = 0.0f;
#pragma unroll
    for (int s = 0; s < 2; ++s) {
      const int  j      = (s == 0) ? jj.x : jj.y;
      const bool padded = (((s == 0) ? (mm & 0xffu) : (mm >> 8)) != 0u);
      const size_t j3 = (size_t)j * 3;
      const float dx = coord[j3 + 0] - xi;
      const float dy = coord[j3 + 1] - yi;
      const float dz = coord[j3 + 2] - zi;
      float d2 = fmaf(dx, dx, fmaf(dy, dy, dz * dz));
      d2 = padded ? 1.0f : d2;                      // masked_fill(pad, 1.0)
      const float d = __builtin_amdgcn_sqrtf(d2);
      float qj = charges[j];
      qj = (padded || (d > Rc)) ? 0.0f : qj;        // kill pad / beyond cutoff
      const float invd = __builtin_amdgcn_rcpf(d);  // rcp(0)=+inf matches ref
      const float c1 = erfc_fast(ALPHA_F * d) * invd;
      acc = fmaf(qi * qj, fmaf(d - Rc, c34, c1 - c2), acc);
    }

    // wave32 all-lane butterfly reduction via ds_swizzle (EXEC full here:
    // loop bounds are wave-uniform, no divergence before this point)
    acc += __int_as_float(__builtin_amdgcn_ds_swizzle(__float_as_int(acc), 0x401F)); // xor 16
    acc += __int_as_float(__builtin_amdgcn_ds_swizzle(__float_as_int(acc), 0x201F)); // xor 8
    acc += __int_as_float(__builtin_amdgcn_ds_swizzle(__float_as_int(acc), 0x101F)); // xor 4
    acc += __int_as_float(__builtin_amdgcn_ds_swizzle(__float_as_int(acc), 0x081F)); // xor 2
    acc += __int_as_float(__builtin_amdgcn_ds_swizzle(__float_as_int(acc), 0x041F)); // xor 1

    if (lane == 0) out[a] = COUL_CONST_F * acc;
  }
}

extern "C" void kernel_launch(void* const* d_in, const int* in_sizes, int n_in,
                              void* d_out, int out_size, void* d_ws, size_t ws_size,
                              hipStream_t stream) {
  (void)n_in; (void)out_size; (void)d_ws; (void)ws_size;
  const float*         coord   = (const float*)d_in[0];
  const float*         charges = (const float*)d_in[1];
  const int*           idxj    = (const int*)d_in[2];
  const unsigned char* mask    = (const unsigned char*)d_in[3];
  const float*         cutoff  = (const float*)d_in[4];
  float*               out     = (float*)d_out;

  const int n_atoms      = in_sizes[1];  // charges is [N]
  const int waves_needed = (n_atoms + ATOMS_PER_WAVE - 1) / ATOMS_PER_WAVE;
  const int blocks       = (waves_needed + WAVES_PER_BLOCK - 1) / WAVES_PER_BLOCK;
  const int waves_total  = blocks * WAVES_PER_BLOCK;

  coul_dsf_kernel<<<blocks, BLOCK_THREADS, 0, stream>>>(
      coord, charges, idxj, mask, cutoff, out, n_atoms, waves_total);
}